// GCNEncoder_83494164234348
// MI455X (gfx1250) — compile-verified
//
#include <hip/hip_runtime.h>
#include <hip/hip_bf16.h>

// ---------------------------------------------------------------------------
// 2-layer GCN (PyG GCNConv semantics) for MI455X / gfx1250.
//   deg[i]   = 1 + sum_{e: col[e]==i} ew[e]          (self-loop weight 1)
//   dis[i]   = rsqrt(deg[i])
//   h        = X @ W                                  (fp32 WMMA 16x16x4)
//   out[i]   = dis[i]^2 * h[i] + b  (self loop seed)
//   out[col] += dis[row]*ew*dis[col] * h[row]          (atomic scatter)
// Layer 1 output gets ReLU (folded into layer-2 GEMM A-load).
// ---------------------------------------------------------------------------

typedef float v2f __attribute__((ext_vector_type(2)));
typedef float v8f __attribute__((ext_vector_type(8)));

#define N_NODES 50000
#define FDIM 128

// ---------------- degree / normalization ----------------

__global__ void __launch_bounds__(256) k_init_deg(float* __restrict__ deg, int n) {
    int i = blockIdx.x * 256 + threadIdx.x;
    if (i < n) deg[i] = 1.0f;   // self-loop weight
}

__global__ void __launch_bounds__(256) k_accum_deg(const int* __restrict__ col,
                                                   const float* __restrict__ ew,
                                                   float* __restrict__ deg, int nE) {
    int e = blockIdx.x * 256 + threadIdx.x;
    if (e < nE) atomicAdd(&deg[col[e]], ew[e]);
}

__global__ void __launch_bounds__(256) k_deg_to_dis(float* __restrict__ deg, int n) {
    int i = blockIdx.x * 256 + threadIdx.x;
    if (i < n) {
        float d = deg[i];
        deg[i] = (d > 0.0f) ? rsqrtf(d) : 0.0f;
    }
}

// ---------------- fp32 WMMA GEMM:  H[50000x128] = X[50000x128] @ W[128x128] --
// One wave32 per 16x16 C tile. blockDim=256 (8 waves): block covers 16 rows x
// full 128 output columns; grid.x = 50000/16 = 3125 tiles in M.
//
// V_WMMA_F32_16X16X4_F32 operand layouts (ISA 7.12.2):
//   A (16x4):  lane L<16 -> row M=L, v[0]=K0, v[1]=K1 ; lane>=16 -> K2,K3
//   B (4x16):  lane L&15 -> col N,   half-wave selects K pair (mirrors A)
//   C (16x16): VGPR j: lanes 0-15 -> M=j, lanes 16-31 -> M=j+8, N = lane&15
template <bool RELU_IN>
__global__ void __launch_bounds__(256) k_gemm_wmma(const float* __restrict__ X,
                                                   const float* __restrict__ W,
                                                   float* __restrict__ H) {
    const int lane = threadIdx.x & 31;
    const int wave = threadIdx.x >> 5;
    const int m0   = blockIdx.x * 16;
    const int n0   = wave * 16;
    const int mrow = m0 + (lane & 15);          // A row this lane feeds
    const int koff = (lane >> 4) * 2;           // half-wave K pair offset
    const int ncol = n0 + (lane & 15);          // B/C column this lane feeds

    v8f c = {};
#pragma unroll
    for (int k = 0; k < FDIM; k += 4) {
        // A fragment: contiguous float2 at (mrow, k+koff)
        const float* ap = X + (size_t)mrow * FDIM + (k + koff);
        float a0 = ap[0];
        float a1 = ap[1];
        if (RELU_IN) { a0 = fmaxf(a0, 0.0f); a1 = fmaxf(a1, 0.0f); }
        v2f a; a[0] = a0; a[1] = a1;
        // B fragment: W[(k+koff)..(k+koff+1)][ncol]
        v2f b;
        b[0] = W[(size_t)(k + koff) * FDIM + ncol];
        b[1] = W[(size_t)(k + koff + 1) * FDIM + ncol];
        c = __builtin_amdgcn_wmma_f32_16x16x4_f32(
                /*neg_a=*/false, a, /*neg_b=*/false, b,
                /*c_mod=*/(short)0, c, /*reuse_a=*/false, /*reuse_b=*/false);
    }

    const int mbase = m0 + 8 * (lane >> 4);
#pragma unroll
    for (int j = 0; j < 8; ++j)
        H[(size_t)(mbase + j) * FDIM + ncol] = c[j];
}

// ---------------- self-loop seed:  OUT[i] = dis[i]^2 * H[i] + bias ----------
__global__ void __launch_bounds__(256) k_seed(const float* __restrict__ H,
                                              const float* __restrict__ dis,
                                              const float* __restrict__ bias,
                                              float* __restrict__ OUT, int nN) {
    int t = blockIdx.x * 256 + threadIdx.x;     // one thread per (node, 4 dims)
    int node = t >> 5;
    int c4   = t & 31;
    if (node >= nN) return;
    float s = dis[node];
    s = s * s;                                  // 1/deg
    float4 h = ((const float4*)(H + (size_t)node * FDIM))[c4];
    float4 b = ((const float4*)bias)[c4];
    float4 o;
    o.x = fmaf(h.x, s, b.x);
    o.y = fmaf(h.y, s, b.y);
    o.z = fmaf(h.z, s, b.z);
    o.w = fmaf(h.w, s, b.w);
    ((float4*)(OUT + (size_t)node * FDIM))[c4] = o;
}

// ---------------- edge scatter:  OUT[col] += norm * H[row] ------------------
// One wave32 per edge; each lane owns a float4 chunk of the 128-dim feature.
// Edge scalars are wave-uniform -> compiler scalarizes to s_loads.
__global__ void __launch_bounds__(256) k_scatter(const int* __restrict__ row,
                                                 const int* __restrict__ col,
                                                 const float* __restrict__ ew,
                                                 const float* __restrict__ dis,
                                                 const float* __restrict__ H,
                                                 float* __restrict__ OUT, int nE) {
    int e = blockIdx.x * 8 + (threadIdx.x >> 5);
    if (e >= nE) return;
    const int lane = threadIdx.x & 31;
    const int r = row[e];
    const int c = col[e];
    const float w = dis[r] * ew[e] * dis[c];
    float4 v = ((const float4*)(H + (size_t)r * FDIM))[lane];
    float* o = OUT + (size_t)c * FDIM + lane * 4;
    atomicAdd(o + 0, v.x * w);
    atomicAdd(o + 1, v.y * w);
    atomicAdd(o + 2, v.z * w);
    atomicAdd(o + 3, v.w * w);
}

// ---------------------------------------------------------------------------

extern "C" void kernel_launch(void* const* d_in, const int* in_sizes, int n_in,
                              void* d_out, int out_size, void* d_ws, size_t ws_size,
                              hipStream_t stream) {
    const float* x  = (const float*)d_in[0];
    const int*   ei = (const int*)d_in[1];     // [2, E]
    const float* ew = (const float*)d_in[2];
    const float* W1 = (const float*)d_in[3];
    const float* b1 = (const float*)d_in[4];
    const float* W2 = (const float*)d_in[5];
    const float* b2 = (const float*)d_in[6];
    float* out = (float*)d_out;

    const int nE = in_sizes[2];                // 1,600,000
    const int nN = in_sizes[0] / FDIM;         // 50,000
    const int* row = ei;
    const int* col = ei + nE;

    // Workspace layout: dis[nN] | bufA[nN*128] | bufB[nN*128]  (~51.4 MB)
    char* ws = (char*)d_ws;
    float* dis  = (float*)ws;
    size_t disBytes = (((size_t)nN * 4) + 255) & ~(size_t)255;
    float* bufA = (float*)(ws + disBytes);
    float* bufB = bufA + (size_t)nN * FDIM;

    const int nodeBlocks = (nN + 255) / 256;
    const int edgeBlocks = (nE + 255) / 256;
    const int seedBlocks = (nN * 32 + 255) / 256;
    const int scatBlocks = (nE + 7) / 8;
    const int gemmBlocks = nN / 16;            // 50000/16 = 3125 exact

    // normalization
    k_init_deg  <<<nodeBlocks, 256, 0, stream>>>(dis, nN);
    k_accum_deg <<<edgeBlocks, 256, 0, stream>>>(col, ew, dis, nE);
    k_deg_to_dis<<<nodeBlocks, 256, 0, stream>>>(dis, nN);

    // layer 1: h1 = x @ W1 ; agg1 = D^-1/2 A_hat D^-1/2 h1 + b1
    k_gemm_wmma<false><<<gemmBlocks, 256, 0, stream>>>(x, W1, bufA);
    k_seed   <<<seedBlocks, 256, 0, stream>>>(bufA, dis, b1, bufB, nN);
    k_scatter<<<scatBlocks, 256, 0, stream>>>(row, col, ew, dis, bufA, bufB, nE);

    // layer 2: h2 = relu(agg1) @ W2 ; out = D^-1/2 A_hat D^-1/2 h2 + b2
    k_gemm_wmma<true><<<gemmBlocks, 256, 0, stream>>>(bufB, W2, bufA);
    k_seed   <<<seedBlocks, 256, 0, stream>>>(bufA, dis, b2, out, nN);
    k_scatter<<<scatBlocks, 256, 0, stream>>>(row, col, ew, dis, bufA, out, nE);
}